// SlidingWindowKVCache_13932873908528
// MI455X (gfx1250) — compile-verified
//
#include <hip/hip_runtime.h>
#include <stdint.h>

#define WINDOW 4096u
#define WMASK  4095u
#define ROWS_PER_BLOCK 128u  // rows (512B each) per block
#define ITERS 16u            // ROWS_PER_BLOCK / (256 threads / 32 lanes-per-row)

// Raw 32-bit LDS byte address from a __shared__ pointer (addrspace(3) ptrtoint,
// NOT the generic-aperture address).
__device__ __forceinline__ unsigned lds_byte_addr(const void* p) {
    return (unsigned)(size_t)(__attribute__((address_space(3))) const void*)p;
}

__device__ __forceinline__ void wait_asynccnt0() {
#if __has_builtin(__builtin_amdgcn_s_wait_asynccnt)
    __builtin_amdgcn_s_wait_asynccnt(0);
#else
    asm volatile("s_wait_asynccnt 0x0" ::: "memory");
#endif
}

// Async-op INST_OFFSET is added to BOTH the LDS and global addresses
// (CDNA5 ISA). Memory stride per iteration (8 rows x
// 512B) == LDS stage stride (256 threads x 16B) == 4096B, so affine runs need
// only immediate offsets.
#define ALOAD(OFF) "global_load_async_to_lds_b128 %0, %1, off offset:" #OFF "\n\t"
#define ASTORE(OFF) "global_store_async_from_lds_b128 %0, %1, off offset:" #OFF "\n\t"

// Output tuple (k_window, v_window) flat in d_out: first N4 float4s = k, rest = v.
// Grid: (ceil(L/128), BH, 2). Block: 256 threads; wave = one 512B row (D=128 fp32).
__global__ void __launch_bounds__(256)
sliding_window_kv_gather(const float4* __restrict__ k_states,
                         const float4* __restrict__ v_states,
                         const float4* __restrict__ k_cache,
                         const float4* __restrict__ v_cache,
                         const int*    __restrict__ p_window_start,
                         const int*    __restrict__ p_total_seq_len,
                         float4*       __restrict__ out,
                         unsigned S,      // seq_len of new states
                         unsigned L,      // output window length
                         unsigned N4)     // float4s per tensor half
{
    static_assert(ITERS == 16u, "asm blobs hardcode 16 iterations");
    __shared__ float4 stage[ITERS][256];

    const unsigned tid  = threadIdx.x;
    const unsigned d4   = tid & 31u;                             // float4 in row
    const unsigned wrow = __builtin_amdgcn_readfirstlane(tid >> 5); // scalar
    const unsigned tile = blockIdx.x;
    const unsigned bh   = blockIdx.y;
    const unsigned half = blockIdx.z;    // 0 = k, 1 = v

    // Scalars live in device memory (graph-capture safe): uniform loads.
    unsigned ws = (unsigned)p_window_start[0] & WMASK;
    unsigned ts = (unsigned)p_total_seq_len[0];

    // Reference logic:
    //   upd   = (ws + ts) % W         (ring base of freshly written states)
    //   start = (ts + S > W) ? (ws + S) % W : ws
    unsigned upd   = (ws + ts) & WMASK;
    unsigned start = (ts + S > WINDOW) ? ((ws + S) & WMASK) : ws;

    const float4* st = half ? v_states : k_states;
    const float4* ca = half ? v_cache  : k_cache;

    // Scalar 64-bit row bases (512 bytes per row).
    unsigned long long newRow0 = (unsigned long long)(size_t)st
                               + (unsigned long long)bh * S      * 512ull;
    unsigned long long oldRow0 = (unsigned long long)(size_t)ca
                               + (unsigned long long)bh * WINDOW * 512ull;
    unsigned long long dstRow0 = (unsigned long long)(size_t)out
                               + (unsigned long long)half * N4   * 16ull
                               + (unsigned long long)bh   * L    * 512ull;

    const unsigned s0      = tile * ROWS_PER_BLOCK + wrow;  // scalar
    const unsigned laneOff = d4 * 16u;                      // per-lane byte off
    const unsigned lds0    = lds_byte_addr(&stage[0][tid]);

    // Affine-run classification (wave-uniform scalars). Rows stride +8 per
    // iteration; last row is s0 + 120.
    const unsigned p0 = (start + s0) & WMASK;
    const unsigned i0 = (p0 - upd) & WMASK;
    const bool tail_ok = (s0 + 8u * (ITERS - 1u)) < L;
    const bool allNew  = (i0 + 8u * (ITERS - 1u)) < S;   // implies i0 < S, no wrap
    const bool allOld  = (i0 >= S) && (i0 + 120u <= WMASK) && (p0 + 120u <= WMASK);
    const bool fast    = tail_ok && (allNew || allOld);

    if (fast) {
        // ---- Fast path: one base address, 16 immediate-offset async loads.
        unsigned long long base = allNew
            ? (newRow0 + (unsigned long long)i0 * 512ull)
            : (oldRow0 + (unsigned long long)p0 * 512ull);
        unsigned long long ga = base + laneOff;       // single 64-bit VALU add
        asm volatile(
            ALOAD(0)     ALOAD(4096)  ALOAD(8192)  ALOAD(12288)
            ALOAD(16384) ALOAD(20480) ALOAD(24576) ALOAD(28672)
            ALOAD(32768) ALOAD(36864) ALOAD(40960) ALOAD(45056)
            ALOAD(49152) ALOAD(53248) ALOAD(57344) ALOAD(61440)
            :: "v"(lds0), "v"(ga) : "memory");
    } else {
        // ---- General path: per-iteration ring/select math (boundary waves).
        #pragma unroll
        for (unsigned it = 0; it < ITERS; ++it) {
            unsigned s = s0 + it * 8u;
            if (s < L) {
                unsigned p = (start + s) & WMASK;
                unsigned i = (p - upd) & WMASK;
                unsigned long long rowBase = (i < S)
                    ? (newRow0 + (unsigned long long)i * 512ull)
                    : (oldRow0 + (unsigned long long)p * 512ull);
                unsigned long long ga = rowBase + laneOff;
                unsigned lds = lds0 + it * 4096u;
                asm volatile("global_load_async_to_lds_b128 %0, %1, off"
                             :: "v"(lds), "v"(ga) : "memory");
            }
        }
    }

    wait_asynccnt0();  // all loads landed in this wave's LDS slots

    if (fast) {
        unsigned long long gd = dstRow0 + (unsigned long long)s0 * 512ull + laneOff;
        asm volatile(
            ASTORE(0)     ASTORE(4096)  ASTORE(8192)  ASTORE(12288)
            ASTORE(16384) ASTORE(20480) ASTORE(24576) ASTORE(28672)
            ASTORE(32768) ASTORE(36864) ASTORE(40960) ASTORE(45056)
            ASTORE(49152) ASTORE(53248) ASTORE(57344) ASTORE(61440)
            :: "v"(gd), "v"(lds0) : "memory");
    } else {
        #pragma unroll
        for (unsigned it = 0; it < ITERS; ++it) {
            unsigned s = s0 + it * 8u;
            if (s < L) {
                unsigned long long gd = dstRow0
                                      + (unsigned long long)s * 512ull + laneOff;
                unsigned lds = lds0 + it * 4096u;
                asm volatile("global_store_async_from_lds_b128 %0, %1, off"
                             :: "v"(gd), "v"(lds) : "memory");
            }
        }
    }

    wait_asynccnt0();  // stores issued to memory before wave retires
}

extern "C" void kernel_launch(void* const* d_in, const int* in_sizes, int n_in,
                              void* d_out, int out_size, void* d_ws, size_t ws_size,
                              hipStream_t stream) {
    const float4* k_states = (const float4*)d_in[0];
    const float4* v_states = (const float4*)d_in[1];
    const float4* k_cache  = (const float4*)d_in[2];
    const float4* v_cache  = (const float4*)d_in[3];
    const int* p_ws = (const int*)d_in[4];
    const int* p_ts = (const int*)d_in[5];
    float4* out = (float4*)d_out;

    const unsigned D = 128, W = 4096;
    unsigned BH = (unsigned)in_sizes[2] / (W * D);      // batch*heads (64)
    unsigned S  = (unsigned)in_sizes[0] / (BH * D);     // new seq len (1024)
    unsigned L  = (unsigned)out_size / (2u * BH * D);   // window length (3072)
    unsigned N4 = (unsigned)out_size / 8u;              // float4s per half

    unsigned nTiles = (L + ROWS_PER_BLOCK - 1u) / ROWS_PER_BLOCK;  // 24

    dim3 grid(nTiles, BH, 2);   // (24, 64, 2) = 3072 blocks, 24576 waves
    dim3 block(256);

    sliding_window_kv_gather<<<grid, block, 0, stream>>>(
        k_states, v_states, k_cache, v_cache, p_ws, p_ts, out, S, L, N4);
}